// VisionEncoderMambaBlock_91070486544594
// MI455X (gfx1250) — compile-verified
//
#include <hip/hip_runtime.h>
#include <hip/hip_bf16.h>
#include <math.h>

#define B_SZ   8
#define SEQ    1024
#define DIM    256
#define DT_RANK 16
#define D_STATE 16
#define NT     256          // NUM_TOKENS
#define POOL   (SEQ / NT)   // 4
#define MTOK   (B_SZ * SEQ) // 8192 tokens

typedef __attribute__((ext_vector_type(2))) float v2f;
typedef __attribute__((ext_vector_type(8))) float v8f;

__device__ __forceinline__ float softplus_f(float x) {
    return (x > 20.f) ? x : log1pf(__expf(x));
}
__device__ __forceinline__ float silu_f(float x) {
    return x / (1.f + __expf(-x));
}

// ---------------------------------------------------------------------------
// LayerNorm over DIM; one token per 256-thread block.
// ---------------------------------------------------------------------------
__global__ __launch_bounds__(256) void ln_kernel(
    const float* __restrict__ x, const float* __restrict__ g,
    const float* __restrict__ b, float* __restrict__ xn)
{
    __shared__ float red[256];
    const int t = blockIdx.x;          // token index 0..MTOK-1
    const int d = threadIdx.x;         // channel 0..255
    const float v = x[(size_t)t * DIM + d];

    red[d] = v; __syncthreads();
    for (int s = 128; s > 0; s >>= 1) { if (d < s) red[d] += red[d + s]; __syncthreads(); }
    const float mu = red[0] * (1.f / DIM);
    __syncthreads();

    const float c = v - mu;
    red[d] = c * c; __syncthreads();
    for (int s = 128; s > 0; s >>= 1) { if (d < s) red[d] += red[d + s]; __syncthreads(); }
    const float var = red[0] * (1.f / DIM);

    xn[(size_t)t * DIM + d] = c * rsqrtf(var + 1e-5f) * g[d] + b[d];
}

// ---------------------------------------------------------------------------
// WMMA f32 GEMM: OUT[M x N] = A[M x K(lda)] @ W[N x K]^T (+bias)(+act)
// Each wave owns a (16*MW) x (16*NW) macro-tile: A fragments reused across NW
// WMMAs, B fragments across MW WMMAs. act: 0=none, 1=softplus.
// A fragment (16x4 f32): lanes 0-15 row M=lane, K-pair {k,k+1}; lanes 16-31 {k+2,k+3}
// C/D (16x16 f32): VGPR r -> row r (lanes 0-15) / r+8 (lanes 16-31), col = lane&15
// ---------------------------------------------------------------------------
template<int MW, int NW>
__global__ __launch_bounds__(256) void wmma_gemm_kernel(
    const float* __restrict__ A, int lda,
    const float* __restrict__ W,       // N x K, row-major
    const float* __restrict__ bias,    // N or nullptr
    float* __restrict__ OUT,           // M x N
    int M, int N, int K, int act)
{
    const int lane = threadIdx.x & 31;
    const int wave = threadIdx.x >> 5;
    const int hsel = lane >> 4;        // 0 or 1 (K half selector)
    const int l16  = lane & 15;
    const int ntiles = N / (16 * NW);
    const int tiles  = (M / (16 * MW)) * ntiles;

    for (int tile = blockIdx.x * 8 + wave; tile < tiles; tile += gridDim.x * 8) {
        const int mt = tile / ntiles, nt = tile - mt * ntiles;
        const int m0 = mt * 16 * MW, n0 = nt * 16 * NW;

        const float* __restrict__ Arow[MW];
        const float* __restrict__ Wrow[NW];
        #pragma unroll
        for (int i = 0; i < MW; ++i) Arow[i] = A + (size_t)(m0 + 16 * i + l16) * lda;
        #pragma unroll
        for (int j = 0; j < NW; ++j) Wrow[j] = W + (size_t)(n0 + 16 * j + l16) * K;

        v8f c[MW][NW];
        #pragma unroll
        for (int i = 0; i < MW; ++i)
            #pragma unroll
            for (int j = 0; j < NW; ++j)
                c[i][j] = (v8f){0.f, 0.f, 0.f, 0.f, 0.f, 0.f, 0.f, 0.f};

        for (int k = 0; k < K; k += 4) {
            const int kk = k + 2 * hsel;
            v2f a[MW], b[NW];
            #pragma unroll
            for (int i = 0; i < MW; ++i) { a[i].x = Arow[i][kk]; a[i].y = Arow[i][kk + 1]; }
            #pragma unroll
            for (int j = 0; j < NW; ++j) { b[j].x = Wrow[j][kk]; b[j].y = Wrow[j][kk + 1]; }
            #pragma unroll
            for (int i = 0; i < MW; ++i)
                #pragma unroll
                for (int j = 0; j < NW; ++j)
                    c[i][j] = __builtin_amdgcn_wmma_f32_16x16x4_f32(
                                  false, a[i], false, b[j], (short)0, c[i][j],
                                  false, false);
        }

        #pragma unroll
        for (int j = 0; j < NW; ++j) {
            const int col = n0 + 16 * j + l16;
            const float bcol = bias ? bias[col] : 0.f;
            #pragma unroll
            for (int i = 0; i < MW; ++i) {
                #pragma unroll
                for (int r = 0; r < 8; ++r) {
                    float v = c[i][j][r] + bcol;
                    if (act == 1) v = softplus_f(v);
                    OUT[(size_t)(m0 + 16 * i + r + 8 * hsel) * N + col] = v;
                }
            }
        }
    }
}

// ---------------------------------------------------------------------------
// Gaussian decay mask (NT weights), computed exactly as the reference.
// ---------------------------------------------------------------------------
__global__ __launch_bounds__(NT) void mask_kernel(float* __restrict__ mask)
{
    __shared__ float red[NT];
    const int i = threadIdx.x;
    const float c = (float)((NT + 1) / 2);
    const float di = (float)i - c;

    red[i] = fabsf(di); __syncthreads();
    for (int s = NT / 2; s > 0; s >>= 1) { if (i < s) red[i] += red[i + s]; __syncthreads(); }
    const float sigma = red[0] * (1.f / NT);
    __syncthreads();

    const float w = __expf(-0.5f * di * di / (sigma * sigma));
    red[i] = w; __syncthreads();
    for (int s = NT / 2; s > 0; s >>= 1) { if (i < s) red[i] += red[i + s]; __syncthreads(); }
    mask[i] = w / red[0];
}

// ---------------------------------------------------------------------------
// Selective scan, one block per batch; dir=0 forward, dir=1 backward (descending t,
// equivalent to flip -> scan -> flip since all projections are per-token).
// Barrier-free: each wave of 32 lanes handles 32 channels; the 32 B/C values for
// the current token are loaded one-per-lane and broadcast via __shfl (ds_bpermute),
// so every wave progresses independently through the recurrence.
// Fused epilogue: 4-wide adaptive avg pool * gaussian mask -> SiLU.
// ---------------------------------------------------------------------------
#define SCAN_PF 8   // prefetch distance (steps)

__global__ __launch_bounds__(DIM) void scan_kernel(
    const float* __restrict__ U,      // (B,SEQ,DIM) ssm input (post conv softplus)
    const float* __restrict__ DELTA,  // (B,SEQ,DIM)
    const float* __restrict__ DBC,    // (B,SEQ,48): [0:16)=dt_pre, [16:32)=B, [32:48)=C
    const float* __restrict__ A_log,  // (DIM, D_STATE)
    const float* __restrict__ Dvec,   // (DIM)
    const float* __restrict__ mask,   // (NT)
    float* __restrict__ XOUT,         // (B, NT, DIM) pooled+masked+silu output
    int dir)
{
    const int b = blockIdx.x;
    const int d = threadIdx.x;         // channel 0..255
    const int lane = threadIdx.x & 31;

    float a[D_STATE], h[D_STATE];
    #pragma unroll
    for (int n = 0; n < D_STATE; ++n) {
        a[n] = -__expf(A_log[d * D_STATE + n]);
        h[n] = 0.f;
    }
    const float Dd = Dvec[d];
    float acc = 0.f;

    for (int step = 0; step < SEQ; ++step) {
        const int t = dir ? (SEQ - 1 - step) : step;
        const size_t tok = (size_t)b * SEQ + t;

        if (step + SCAN_PF < SEQ) {
            const int tp = dir ? (SEQ - 1 - (step + SCAN_PF)) : (step + SCAN_PF);
            const size_t tokp = (size_t)b * SEQ + tp;
            __builtin_prefetch(&U[tokp * DIM + d], 0, 1);
            __builtin_prefetch(&DELTA[tokp * DIM + d], 0, 1);
            __builtin_prefetch(&DBC[tokp * 48 + DT_RANK + lane], 0, 1);
        }

        const float u     = U[tok * DIM + d];
        const float delta = DELTA[tok * DIM + d];
        // lane 0..15 -> B[0..15], lane 16..31 -> C[0..15] (contiguous in DBC row)
        const float bc = DBC[tok * 48 + DT_RANK + lane];

        const float du = delta * u;
        float y = 0.f;
        #pragma unroll
        for (int n = 0; n < D_STATE; ++n) {
            const float Bn = __shfl(bc, n, 32);
            const float Cn = __shfl(bc, 16 + n, 32);
            h[n] = __expf(delta * a[n]) * h[n] + du * Bn;
            y += h[n] * Cn;
        }
        y += u * Dd;
        acc += y;

        if ((step & (POOL - 1)) == (POOL - 1)) {
            const int p = t >> 2;                  // pooled token (window aligned both dirs)
            const float pv = acc * (1.f / POOL) * mask[p];
            XOUT[((size_t)b * NT + p) * DIM + d] = silu_f(pv);
            acc = 0.f;
        }
    }
}

// ---------------------------------------------------------------------------
// Final combine: out = silu(pool(z)) * (x1 + x2) + pool(x)
// ---------------------------------------------------------------------------
__global__ __launch_bounds__(256) void combine_kernel(
    const float* __restrict__ x, const float* __restrict__ z,
    const float* __restrict__ x1, const float* __restrict__ x2,
    float* __restrict__ out)
{
    const int i = blockIdx.x * 256 + threadIdx.x;     // over B*NT*DIM
    const int d = i & (DIM - 1);
    const int p = (i >> 8) & (NT - 1);
    const int b = i >> 16;
    const size_t base = ((size_t)b * SEQ + p * POOL) * DIM + d;

    float zp = 0.f, sk = 0.f;
    #pragma unroll
    for (int j = 0; j < POOL; ++j) {
        zp += z[base + (size_t)j * DIM];
        sk += x[base + (size_t)j * DIM];
    }
    zp *= (1.f / POOL);
    sk *= (1.f / POOL);
    zp = silu_f(zp);
    out[i] = zp * (x1[i] + x2[i]) + sk;
}

// ---------------------------------------------------------------------------
// Launch
// ---------------------------------------------------------------------------
static inline int gemm_blocks(int M, int N, int MW, int NW) {
    int tiles = (M / (16 * MW)) * (N / (16 * NW));
    int blk = (tiles + 7) / 8;
    return blk > 2048 ? 2048 : (blk < 1 ? 1 : blk);
}

extern "C" void kernel_launch(void* const* d_in, const int* in_sizes, int n_in,
                              void* d_out, int out_size, void* d_ws, size_t ws_size,
                              hipStream_t stream) {
    const float* x      = (const float*)d_in[0];
    const float* ln_g   = (const float*)d_in[1];
    const float* ln_b   = (const float*)d_in[2];
    const float* p1_w   = (const float*)d_in[3];
    const float* p1_b   = (const float*)d_in[4];
    const float* p2_w   = (const float*)d_in[5];
    const float* p2_b   = (const float*)d_in[6];
    const float* cf_w   = (const float*)d_in[7];
    const float* cf_b   = (const float*)d_in[8];
    const float* cb_w   = (const float*)d_in[9];
    const float* cb_b   = (const float*)d_in[10];
    const float* f_dbc_w = (const float*)d_in[11];
    const float* f_dt_w  = (const float*)d_in[12];
    const float* f_dt_b  = (const float*)d_in[13];
    const float* f_A_log = (const float*)d_in[14];
    const float* f_D     = (const float*)d_in[15];
    const float* b_dbc_w = (const float*)d_in[16];
    const float* b_dt_w  = (const float*)d_in[17];
    const float* b_dt_b  = (const float*)d_in[18];
    const float* b_A_log = (const float*)d_in[19];
    const float* b_D     = (const float*)d_in[20];
    float* out = (float*)d_out;

    // workspace layout (floats)
    float* ws = (float*)d_ws;
    const size_t MD = (size_t)MTOK * DIM;   // 2,097,152
    float* xn      = ws;               ws += MD;
    float* z       = ws;               ws += MD;
    float* xp      = ws;               ws += MD;
    float* uf      = ws;               ws += MD;
    float* ub      = ws;               ws += MD;
    float* f_delta = ws;               ws += MD;
    float* b_delta = ws;               ws += MD;
    float* f_dbc   = ws;               ws += (size_t)MTOK * 48;
    float* b_dbc   = ws;               ws += (size_t)MTOK * 48;
    float* x1      = ws;               ws += (size_t)B_SZ * NT * DIM;
    float* x2      = ws;               ws += (size_t)B_SZ * NT * DIM;
    float* gmask   = ws;               ws += NT;

    // 1) LayerNorm
    ln_kernel<<<MTOK, 256, 0, stream>>>(x, ln_g, ln_b, xn);

    // 2) Projections (WMMA f32 GEMMs, 32x64 macro-tile per wave)
    wmma_gemm_kernel<2, 4><<<gemm_blocks(MTOK, DIM, 2, 4), 256, 0, stream>>>(
        xn, DIM, p1_w, p1_b, z, MTOK, DIM, DIM, 0);
    wmma_gemm_kernel<2, 4><<<gemm_blocks(MTOK, DIM, 2, 4), 256, 0, stream>>>(
        xn, DIM, p2_w, p2_b, xp, MTOK, DIM, DIM, 0);
    // conv1d(k=1) + softplus for both directions (per-token, flip handled in scan)
    wmma_gemm_kernel<2, 4><<<gemm_blocks(MTOK, DIM, 2, 4), 256, 0, stream>>>(
        xp, DIM, cf_w, cf_b, uf, MTOK, DIM, DIM, 1);
    wmma_gemm_kernel<2, 4><<<gemm_blocks(MTOK, DIM, 2, 4), 256, 0, stream>>>(
        xp, DIM, cb_w, cb_b, ub, MTOK, DIM, DIM, 1);
    // dbc projections: (M x 256) @ (48 x 256)^T, 32x48 macro-tile
    wmma_gemm_kernel<2, 3><<<gemm_blocks(MTOK, 48, 2, 3), 256, 0, stream>>>(
        uf, DIM, f_dbc_w, nullptr, f_dbc, MTOK, 48, DIM, 0);
    wmma_gemm_kernel<2, 3><<<gemm_blocks(MTOK, 48, 2, 3), 256, 0, stream>>>(
        ub, DIM, b_dbc_w, nullptr, b_dbc, MTOK, 48, DIM, 0);
    // delta projections: softplus((M x 16 slice of dbc, lda=48) @ (256 x 16)^T + b)
    wmma_gemm_kernel<2, 4><<<gemm_blocks(MTOK, DIM, 2, 4), 256, 0, stream>>>(
        f_dbc, 48, f_dt_w, f_dt_b, f_delta, MTOK, DIM, DT_RANK, 1);
    wmma_gemm_kernel<2, 4><<<gemm_blocks(MTOK, DIM, 2, 4), 256, 0, stream>>>(
        b_dbc, 48, b_dt_w, b_dt_b, b_delta, MTOK, DIM, DT_RANK, 1);

    // 3) Gaussian mask
    mask_kernel<<<1, NT, 0, stream>>>(gmask);

    // 4) Selective scans (fwd + bwd), fused pool*mask -> silu
    scan_kernel<<<B_SZ, DIM, 0, stream>>>(uf, f_delta, f_dbc, f_A_log, f_D, gmask, x1, 0);
    scan_kernel<<<B_SZ, DIM, 0, stream>>>(ub, b_delta, b_dbc, b_A_log, b_D, gmask, x2, 1);

    // 5) Combine
    combine_kernel<<<(B_SZ * NT * DIM) / 256, 256, 0, stream>>>(x, z, x1, x2, out);
}